// SageEmbedder_69870527971697
// MI455X (gfx1250) — compile-verified
//
#include <hip/hip_runtime.h>
#include <hip/hip_bf16.h>

typedef __attribute__((ext_vector_type(2))) float v2f;
typedef __attribute__((ext_vector_type(8))) float v8f;

#define FEAT 128
#define LDS_STRIDE 132   // 132 % 64 == 4 -> rows spread across banks, conflict-free b64 reads

// ---------------------------------------------------------------------------
// degree count: one thread per edge, atomicAdd 1.0f into deg[dst[e]]
// ---------------------------------------------------------------------------
__global__ void sage_degree_kernel(const int* __restrict__ dst,
                                   float* __restrict__ deg, int E) {
    int t = blockIdx.x * blockDim.x + threadIdx.x;
    if (t < E) atomicAdd(&deg[dst[t]], 1.0f);
}

// deg -> inv_deg in place (0 for isolated nodes)
__global__ void sage_invdeg_kernel(float* __restrict__ deg, int N) {
    int t = blockIdx.x * blockDim.x + threadIdx.x;
    if (t < N) {
        float d = deg[t];
        deg[t] = (d > 0.0f) ? (1.0f / d) : 0.0f;
    }
}

// ---------------------------------------------------------------------------
// edge scatter: one wave per edge; 32 lanes x float4 cover the 128 features.
// Coalesced 512B gather of h[src], 4 f32 atomics per lane into neigh[dst].
// ---------------------------------------------------------------------------
__global__ __launch_bounds__(256)
void sage_scatter_kernel(const float* __restrict__ h, const int* __restrict__ src,
                         const int* __restrict__ dst, float* __restrict__ neigh,
                         int E) {
    int t = blockIdx.x * blockDim.x + threadIdx.x;
    int e = t >> 5;
    if (e >= E) return;
    int c = (t & 31) << 2;                       // feature chunk (4 floats)
    const float4 v = *(const float4*)(h + (size_t)src[e] * FEAT + c);
    float* p = neigh + (size_t)dst[e] * FEAT + c;
    atomicAdd(p + 0, v.x);
    atomicAdd(p + 1, v.y);
    atomicAdd(p + 2, v.z);
    atomicAdd(p + 3, v.w);
}

// ---------------------------------------------------------------------------
// fused layer GEMM:  out = h @ Wself + (neigh * inv_deg) @ Wneigh + bias
// 256 threads = 8 waves per block; block computes a 16x128 output tile,
// each wave owns one 16x16 column tile via V_WMMA_F32_16X16X4_F32.
// ---------------------------------------------------------------------------
__global__ __launch_bounds__(256, 2)
void sage_gemm_kernel(const float* __restrict__ hself,
                      const float* __restrict__ neigh,
                      const float* __restrict__ invdeg,
                      const float* __restrict__ Wself,
                      const float* __restrict__ Wneigh,
                      const float* __restrict__ bias,
                      float* __restrict__ out,
                      int nrows, int do_tanh) {
    __shared__ float As[16 * LDS_STRIDE];
    __shared__ float An[16 * LDS_STRIDE];

    const int rbase = blockIdx.x * 16;
    const int tid   = threadIdx.x;

    // Cooperative stage: 16 rows x 128 cols, float4 chunks; scale neigh rows
    // by inv_deg while staging.
    for (int i = tid; i < 512; i += 256) {
        const int row = i >> 5;
        const int c4  = (i & 31) << 2;
        int rr = rbase + row;
        if (rr >= nrows) rr = nrows - 1;          // clamp (N is exact multiple anyway)
        const float4 vs = *(const float4*)(hself + (size_t)rr * FEAT + c4);
        const float4 vn = *(const float4*)(neigh + (size_t)rr * FEAT + c4);
        const float  sc = invdeg[rr];
        float* ps = &As[row * LDS_STRIDE + c4];
        float* pn = &An[row * LDS_STRIDE + c4];
        ps[0] = vs.x; ps[1] = vs.y; ps[2] = vs.z; ps[3] = vs.w;
        pn[0] = vn.x * sc; pn[1] = vn.y * sc; pn[2] = vn.z * sc; pn[3] = vn.w * sc;
    }
    __syncthreads();

    const int wave = tid >> 5;          // column tile 0..7
    const int lane = tid & 31;
    const int n    = lane & 15;         // N (and M for A-frag) index
    const int hi   = lane >> 4;         // half-wave select
    const int koff = hi * 2;            // A/B frag K sub-offset
    const int col  = (wave << 4) + n;

    const float* __restrict__ as = &As[n * LDS_STRIDE + koff];
    const float* __restrict__ an = &An[n * LDS_STRIDE + koff];

    v8f acc = {};
    for (int k = 0; k < FEAT; k += 4) {
        // A frag: lane m holds A[m][k+koff], A[m][k+koff+1]  (b64 LDS load)
        v2f a0; a0.x = as[k]; a0.y = as[k + 1];
        // B frag: lane n holds W[k+koff][col], W[k+koff+1][col]
        v2f b0; b0.x = Wself[(size_t)(k + koff) * FEAT + col];
                b0.y = Wself[(size_t)(k + koff + 1) * FEAT + col];
        acc = __builtin_amdgcn_wmma_f32_16x16x4_f32(false, a0, false, b0,
                                                    (short)0, acc, false, false);
        v2f a1; a1.x = an[k]; a1.y = an[k + 1];
        v2f b1; b1.x = Wneigh[(size_t)(k + koff) * FEAT + col];
                b1.y = Wneigh[(size_t)(k + koff + 1) * FEAT + col];
        acc = __builtin_amdgcn_wmma_f32_16x16x4_f32(false, a1, false, b1,
                                                    (short)0, acc, false, false);
    }

    // C/D layout: vgpr r -> M = r + 8*hi, N = lane&15. Bias depends only on N.
    const float bv = bias[col];
    const int mbase = rbase + hi * 8;
#pragma unroll
    for (int r = 0; r < 8; ++r) {
        const int row = mbase + r;
        if (row < nrows) {
            float v = acc[r] + bv;
            if (do_tanh) v = tanhf(v);
            out[(size_t)row * FEAT + col] = v;
        }
    }
}

// ---------------------------------------------------------------------------
// host-side launch
// inputs: x[N,128], src[E], dst[E], W_self[L,128,128], W_neigh[L,128,128], b[L,128]
// ---------------------------------------------------------------------------
extern "C" void kernel_launch(void* const* d_in, const int* in_sizes, int n_in,
                              void* d_out, int out_size, void* d_ws, size_t ws_size,
                              hipStream_t stream) {
    const float* x       = (const float*)d_in[0];
    const int*   src     = (const int*)d_in[1];
    const int*   dst     = (const int*)d_in[2];
    const float* W_self  = (const float*)d_in[3];
    const float* W_neigh = (const float*)d_in[4];
    const float* b       = (const float*)d_in[5];
    float*       out     = (float*)d_out;

    const int N = in_sizes[0] / FEAT;
    const int E = in_sizes[1];

    // workspace layout: invdeg[N] | neigh[N*128] | h1[N*128]
    char*  ws     = (char*)d_ws;
    float* invdeg = (float*)ws;
    size_t o1     = (((size_t)N * 4 + 255) / 256) * 256;
    float* neigh  = (float*)(ws + o1);
    size_t o2     = o1 + ((((size_t)N * FEAT * 4) + 255) / 256) * 256;
    float* h1     = (float*)(ws + o2);

    const int threads     = 256;
    const int deg_blocks  = (E + threads - 1) / threads;
    const int inv_blocks  = (N + threads - 1) / threads;
    const int scat_blocks = (int)(((size_t)E * 32 + threads - 1) / threads);
    const int gemm_blocks = (N + 15) / 16;

    // degrees
    hipMemsetAsync(invdeg, 0, (size_t)N * sizeof(float), stream);
    sage_degree_kernel<<<deg_blocks, threads, 0, stream>>>(dst, invdeg, E);
    sage_invdeg_kernel<<<inv_blocks, threads, 0, stream>>>(invdeg, N);

    // layer 0: x -> h1
    hipMemsetAsync(neigh, 0, (size_t)N * FEAT * sizeof(float), stream);
    sage_scatter_kernel<<<scat_blocks, threads, 0, stream>>>(x, src, dst, neigh, E);
    sage_gemm_kernel<<<gemm_blocks, threads, 0, stream>>>(
        x, neigh, invdeg, W_self, W_neigh, b, h1, N, 0);

    // layer 1: h1 -> out (with tanh)
    hipMemsetAsync(neigh, 0, (size_t)N * FEAT * sizeof(float), stream);
    sage_scatter_kernel<<<scat_blocks, threads, 0, stream>>>(h1, src, dst, neigh, E);
    sage_gemm_kernel<<<gemm_blocks, threads, 0, stream>>>(
        h1, neigh, invdeg, W_self + FEAT * FEAT, W_neigh + FEAT * FEAT,
        b + FEAT, out, N, 1);
}